// NMSquaredGaussianMixture_15229954031939
// MI455X (gfx1250) — compile-verified
//
#include <hip/hip_runtime.h>
#include <math.h>

typedef __attribute__((ext_vector_type(2))) float v2f;
typedef __attribute__((ext_vector_type(8))) float v8f;

#define N_CLUSTERS 32
#define N_POINTS_C 1000000
#define TILE 16
#define N_TILES (N_POINTS_C / TILE)          // 62500 exactly, no tail
#define MAIN_BLOCKS 512
#define MAIN_THREADS 256
#define WAVES_PER_BLOCK (MAIN_THREADS / 32)  // 8 waves (wave32)
#define TOTAL_WAVES (MAIN_BLOCKS * WAVES_PER_BLOCK)

// workspace layout (floats)
#define WS_COEFF   0     // C[8][32] row-major (rows 6,7 zero pad)
#define WS_WSCALED 256   // w_k / norm_k, 32 floats
#define WS_Z       288   // scalar Z
#define WS_PARTIAL 320   // MAIN_BLOCKS block partials

// DPP row-rotate move (row of 16 lanes), bound_ctrl=1, all rows/banks.
template <int CTRL>
__device__ __forceinline__ float dpp_mov_f32(float x) {
    int xi = __builtin_bit_cast(int, x);
    int r  = __builtin_amdgcn_update_dpp(0, xi, CTRL, 0xF, 0xF, true);
    return __builtin_bit_cast(float, r);
}
// Sum across the 16-lane row via rotate reduction (pure VALU, no LDS).
__device__ __forceinline__ float row16_sum(float v) {
    v += dpp_mov_f32<0x121>(v);  // row_ror:1
    v += dpp_mov_f32<0x122>(v);  // row_ror:2
    v += dpp_mov_f32<0x124>(v);  // row_ror:4
    v += dpp_mov_f32<0x128>(v);  // row_ror:8
    return v;
}

__global__ void __launch_bounds__(1024)
setup_kernel(const float* __restrict__ means,
             const float* __restrict__ chols,
             const float* __restrict__ weights,
             float* __restrict__ ws) {
    __shared__ float sl00[32], sl10[32], sl11[32], smx[32], smy[32], sw[32];
    __shared__ float red[1024];
    const float TWO_PI = 6.283185307179586f;
    int tid = threadIdx.x;
    if (tid < 32) {
        int k = tid;
        // tril(chols[k]) = [[a,0],[b,c]]
        float a = chols[k * 4 + 0];
        float b = chols[k * 4 + 2];
        float c = chols[k * 4 + 3];
        // S = T T^T + I = [[a^2+1, ab],[ab, b^2+c^2+1]]; L = chol(S)
        float l00 = sqrtf(a * a + 1.0f);
        float l10 = a * b / l00;
        float l11 = sqrtf(b * b + c * c + 1.0f - l10 * l10);
        // L^{-1} = [[1/l00,0],[-l10/(l00 l11), 1/l11]]
        float A  = 1.0f / l00;
        float Bq = -l10 / (l00 * l11);
        float Cq = 1.0f / l11;
        float norm = TWO_PI * sqrtf(l00 * l11);  // sqrt((2pi)^2 det L)
        float mx = means[k * 2 + 0], my = means[k * 2 + 1];
        // g = -0.5 * [A(x-mx)^2 + B(x-mx)(y-my) + C(y-my)^2] expanded in
        // basis [x^2, xy, y^2, x, y, 1]; -0.5 folded in.
        ws[WS_COEFF + 0 * 32 + k] = -0.5f * A;
        ws[WS_COEFF + 1 * 32 + k] = -0.5f * Bq;
        ws[WS_COEFF + 2 * 32 + k] = -0.5f * Cq;
        ws[WS_COEFF + 3 * 32 + k] = A * mx + 0.5f * Bq * my;
        ws[WS_COEFF + 4 * 32 + k] = 0.5f * Bq * mx + Cq * my;
        ws[WS_COEFF + 5 * 32 + k] = -0.5f * (A * mx * mx + Bq * mx * my + Cq * my * my);
        ws[WS_COEFF + 6 * 32 + k] = 0.0f;   // K pad
        ws[WS_COEFF + 7 * 32 + k] = 0.0f;   // K pad
        ws[WS_WSCALED + k] = weights[k] / norm;
        sl00[k] = l00; sl10[k] = l10; sl11[k] = l11;
        smx[k] = mx; smy[k] = my; sw[k] = weights[k];
    }
    __syncthreads();
    // Z: all 32x32 (i,j) pairs, one thread each
    int i = tid >> 5, j = tid & 31;
    float s00 = sl00[i] + sl00[j];
    float s10 = sl10[i] + sl10[j];
    float s11 = sl11[i] + sl11[j];
    float dmx = smx[i] - smx[j], dmy = smy[i] - smy[j];
    float qz = dmx * dmx / s00 - (s10 / (s00 * s11)) * dmx * dmy + dmy * dmy / s11;
    float z = __expf(-0.5f * qz) / (TWO_PI * sqrtf(s00 * s11)) * sw[i] * sw[j];
    red[tid] = z;
    __syncthreads();
    for (int off = 512; off > 0; off >>= 1) {
        if (tid < off) red[tid] += red[tid + off];
        __syncthreads();
    }
    if (tid == 0) ws[WS_Z] = red[0];
}

__global__ void __launch_bounds__(MAIN_THREADS)
main_kernel(const float* __restrict__ X,
            const float* __restrict__ ws,
            float* __restrict__ partials) {
    __shared__ float wsum[WAVES_PER_BLOCK];
    const int lane = threadIdx.x & 31;
    const int waveInBlock = threadIdx.x >> 5;
    const int wave = blockIdx.x * WAVES_PER_BLOCK + waveInBlock;
    const int row = lane & 15;   // M index (points) / N index (clusters)
    const int half = lane >> 4;  // selects K pair per 16x4 f32 A layout

    // B operand: columns of C[8][32]; per documented 32-bit operand layout:
    // lanes 0-15 hold K={0,1} rows, lanes 16-31 hold K={2,3} rows (per step).
    const float* C = ws + WS_COEFF;
    const int n0 = row, n1 = 16 + row;  // two 16-cluster column tiles
    v2f b0_t0, b1_t0, b0_t1, b1_t1;
    b0_t0.x = C[(half * 2 + 0) * 32 + n0]; b0_t0.y = C[(half * 2 + 1) * 32 + n0];
    b1_t0.x = C[(half * 2 + 4) * 32 + n0]; b1_t0.y = C[(half * 2 + 5) * 32 + n0];
    b0_t1.x = C[(half * 2 + 0) * 32 + n1]; b0_t1.y = C[(half * 2 + 1) * 32 + n1];
    b1_t1.x = C[(half * 2 + 4) * 32 + n1]; b1_t1.y = C[(half * 2 + 5) * 32 + n1];
    const float wsc0 = ws[WS_WSCALED + n0];
    const float wsc1 = ws[WS_WSCALED + n1];

    const float2* __restrict__ X2 = (const float2*)X;

    float wacc = 0.0f;
    for (int tile = wave; tile < N_TILES; tile += TOTAL_WAVES) {
        const int p = tile * 16 + row;
        const float2 pt = X2[p];         // single global_load_b64
        const float x = pt.x, y = pt.y;
        {   // prefetch our next strided tile (wave-uniform branch)
            int pf = tile + TOTAL_WAVES;
            if (pf < N_TILES) __builtin_prefetch((const void*)(X + (size_t)pf * 32), 0, 0);
        }
        // A operand: Phi = [x^2, xy, y^2, x, y, 1, 0, 0] per 16x4 f32 layout.
        v2f a0, a1;
        a0.x = half ? y * y : x * x;   // K0 / K2
        a0.y = half ? x     : x * y;   // K1 / K3
        a1.x = half ? 0.0f  : y;       // K4 / K6
        a1.y = half ? 0.0f  : 1.0f;    // K5 / K7

        v8f acc0 = {};
        acc0 = __builtin_amdgcn_wmma_f32_16x16x4_f32(false, a0, false, b0_t0,
                                                     (short)0, acc0, false, false);
        acc0 = __builtin_amdgcn_wmma_f32_16x16x4_f32(false, a1, false, b1_t0,
                                                     (short)0, acc0, false, false);
        v8f acc1 = {};
        acc1 = __builtin_amdgcn_wmma_f32_16x16x4_f32(false, a0, false, b0_t1,
                                                     (short)0, acc1, false, false);
        acc1 = __builtin_amdgcn_wmma_f32_16x16x4_f32(false, a1, false, b1_t1,
                                                     (short)0, acc1, false, false);

        // D layout: lane holds (m = r + half*8, n = tile_col*16 + row).
        // s_m = sum_n (w_n/norm_n) * exp(g_{m,n})
        float tilesum = 0.0f;
        #pragma unroll
        for (int r = 0; r < 8; ++r) {
            float t = wsc0 * __expf(acc0[r]) + wsc1 * __expf(acc1[r]);
            float s = row16_sum(t);      // DPP rotate reduction, no LDS
            tilesum += s * s;            // num_m = s_m^2
        }
        wacc += (row == 0) ? tilesum : 0.0f;  // count once per 16-lane group
    }
    wacc += __shfl_xor(wacc, 16);  // combine halves (m=0..7 and m=8..15)
    if (lane == 0) wsum[waveInBlock] = wacc;
    __syncthreads();
    if (threadIdx.x == 0) {
        float b = 0.0f;
        for (int w = 0; w < WAVES_PER_BLOCK; ++w) b += wsum[w];
        partials[blockIdx.x] = b;
    }
}

__global__ void finalize_kernel(const float* __restrict__ partials,
                                const float* __restrict__ ws,
                                float* __restrict__ out) {
    if (threadIdx.x == 0 && blockIdx.x == 0) {
        float S = 0.0f;
        for (int i = 0; i < MAIN_BLOCKS; ++i) S += partials[i];  // fixed order
        float Z = ws[WS_Z];
        // -(logsumexp(log pdf))/N = -(log(sum num) - log Z)/N
        out[0] = -(logf(S) - logf(Z)) / (float)N_POINTS_C;
    }
}

extern "C" void kernel_launch(void* const* d_in, const int* in_sizes, int n_in,
                              void* d_out, int out_size, void* d_ws, size_t ws_size,
                              hipStream_t stream) {
    const float* X       = (const float*)d_in[0];
    const float* means   = (const float*)d_in[1];
    const float* chols   = (const float*)d_in[2];
    const float* weights = (const float*)d_in[3];
    float* ws  = (float*)d_ws;
    float* out = (float*)d_out;

    setup_kernel<<<1, 1024, 0, stream>>>(means, chols, weights, ws);
    main_kernel<<<MAIN_BLOCKS, MAIN_THREADS, 0, stream>>>(X, ws, ws + WS_PARTIAL);
    finalize_kernel<<<1, 64, 0, stream>>>(ws + WS_PARTIAL, ws, out);
}